// PointsRenderer_16406775070833
// MI455X (gfx1250) — compile-verified
//
#include <hip/hip_runtime.h>

// Points alpha-compositing renderer for MI455X (gfx1250).
// Pure HBM-bandwidth-bound kernel (~170 MB traffic, ~0.2 GFLOP -> AI ~1.2).
// Ideal time ~= 170 MB / 23.3 TB/s ~= 7.3 us. WMMA deliberately unused: the
// per-pixel 1x8 . 8x4 contraction has per-pixel gathered operands (no shared
// matrix per wave), so matrix ops cannot apply.
// CDNA5 features used:
//   * global_load_async_to_lds_b128 th:TH_LOAD_NT  (ASYNCcnt async copy,
//     double buffered, streaming data kept out of the L2 feature working set)
//   * s_wait_asynccnt                              (split dependency counter)
//   * non-temporal b128 output stores; features (1.6 MB) stay L2-resident.

#define TILE 256      // pixels per tile == blockDim.x
#define KFRAG 8       // fragments per pixel
#define CFEAT 4       // feature channels

typedef float vfloat4 __attribute__((ext_vector_type(4)));

__device__ __forceinline__ unsigned lds_byte_off(const void* p) {
  // generic pointer to __shared__ carries the LDS offset in its low 32 bits
  return (unsigned)(unsigned long long)p;
}

// Stage one pixel's 8 idx (32B) + 8 dists2 (32B) into LDS asynchronously.
// INST_OFFSET applies to BOTH the global and LDS address, so one base pair
// covers both 16B halves. Non-temporal: this data is consumed exactly once.
__device__ __forceinline__ void stage_async(const int* gI, const float* gD,
                                            const int* lI, const float* lD) {
  unsigned li = lds_byte_off(lI);
  unsigned ld = lds_byte_off(lD);
  unsigned long long gi = (unsigned long long)gI;
  unsigned long long gd = (unsigned long long)gD;
  asm volatile(
      "global_load_async_to_lds_b128 %0, %1, off th:TH_LOAD_NT\n\t"
      "global_load_async_to_lds_b128 %0, %1, off offset:16 th:TH_LOAD_NT\n\t"
      "global_load_async_to_lds_b128 %2, %3, off th:TH_LOAD_NT\n\t"
      "global_load_async_to_lds_b128 %2, %3, off offset:16 th:TH_LOAD_NT"
      :
      : "v"(li), "v"(gi), "v"(ld), "v"(gd)
      : "memory");
}

__device__ __forceinline__ vfloat4 composite8(const int* id, const float* d2,
                                              float inv_r2,
                                              const float4* __restrict__ feat4,
                                              int npts) {
  float T = 1.0f;
  vfloat4 acc = {0.f, 0.f, 0.f, 0.f};
#pragma unroll
  for (int k = 0; k < KFRAG; ++k) {
    const float w = 1.0f - d2[k] * inv_r2;           // weight in [0,1]
    const float a = (id[k] >= 0) ? w : 0.0f;         // invalid slots contribute 0
    int cid = id[k] < 0 ? 0 : id[k];                 // clip like the reference
    cid = cid >= npts ? npts - 1 : cid;              // safety clamp
    const float4 f = feat4[cid];                     // global_load_b128, L2-hot
    const float c = a * T;                           // alpha * transmittance
    acc.x = fmaf(c, f.x, acc.x);
    acc.y = fmaf(c, f.y, acc.y);
    acc.z = fmaf(c, f.z, acc.z);
    acc.w = fmaf(c, f.w, acc.w);
    T = T * (1.0f - a);
  }
  return acc;
}

__global__ __launch_bounds__(TILE) void points_alpha_composite(
    const int* __restrict__ idx,        // [npix, K]
    const float* __restrict__ dists2,   // [npix, K]
    const float* __restrict__ features, // [npts, C]
    const float* __restrict__ radius,   // [1]
    float* __restrict__ out,            // [npix, C]
    int npix, int npts) {
  __shared__ __align__(16) int   sIdx[2][TILE * KFRAG];  // 2 x 8 KB
  __shared__ __align__(16) float sD2[2][TILE * KFRAG];   // 2 x 8 KB

  const int t = (int)threadIdx.x;
  const float r = radius[0];
  const float inv_r2 = 1.0f / (r * r);
  const float4* __restrict__ feat4 = reinterpret_cast<const float4*>(features);
  vfloat4* __restrict__ out4 = reinterpret_cast<vfloat4*>(out);

  const int ntiles = npix / TILE;
  int tile = (int)blockIdx.x;
  int buf = 0;

  if (tile < ntiles) {
    // prologue: stage first tile
    stage_async(idx + (size_t)(tile * TILE + t) * KFRAG,
                dists2 + (size_t)(tile * TILE + t) * KFRAG,
                &sIdx[0][t * KFRAG], &sD2[0][t * KFRAG]);

    for (; tile < ntiles; tile += (int)gridDim.x) {
      const int nxt = tile + (int)gridDim.x;
      if (nxt < ntiles) {
        // stage next tile into the other buffer, then wait for the current
        // tile's 4 async ops (per-wave counter, in-order completion)
        stage_async(idx + (size_t)(nxt * TILE + t) * KFRAG,
                    dists2 + (size_t)(nxt * TILE + t) * KFRAG,
                    &sIdx[buf ^ 1][t * KFRAG], &sD2[buf ^ 1][t * KFRAG]);
        asm volatile("s_wait_asynccnt 4" ::: "memory");
      } else {
        asm volatile("s_wait_asynccnt 0" ::: "memory");
      }

      // each thread consumes exactly what it staged -> no workgroup barrier
      const int4* vI = reinterpret_cast<const int4*>(&sIdx[buf][t * KFRAG]);
      const float4* vD = reinterpret_cast<const float4*>(&sD2[buf][t * KFRAG]);
      const int4 ia = vI[0], ib = vI[1];          // ds_load_b128 x2
      const float4 da = vD[0], db = vD[1];        // ds_load_b128 x2
      const int id[KFRAG] = {ia.x, ia.y, ia.z, ia.w, ib.x, ib.y, ib.z, ib.w};
      const float d2[KFRAG] = {da.x, da.y, da.z, da.w, db.x, db.y, db.z, db.w};

      const vfloat4 acc = composite8(id, d2, inv_r2, feat4, npts);
      // write-once output: non-temporal b128 store
      __builtin_nontemporal_store(acc, &out4[tile * TILE + t]);
      buf ^= 1;
    }
  }

  // generic tail for npix not divisible by TILE (empty for the given sizes):
  for (int p = ntiles * TILE + (int)(blockIdx.x * blockDim.x) + t; p < npix;
       p += (int)(gridDim.x * blockDim.x)) {
    const int4* gI = reinterpret_cast<const int4*>(idx + (size_t)p * KFRAG);
    const float4* gD = reinterpret_cast<const float4*>(dists2 + (size_t)p * KFRAG);
    const int4 ia = gI[0], ib = gI[1];
    const float4 da = gD[0], db = gD[1];
    const int id[KFRAG] = {ia.x, ia.y, ia.z, ia.w, ib.x, ib.y, ib.z, ib.w};
    const float d2[KFRAG] = {da.x, da.y, da.z, da.w, db.x, db.y, db.z, db.w};
    const vfloat4 acc = composite8(id, d2, inv_r2, feat4, npts);
    __builtin_nontemporal_store(acc, &out4[p]);
  }
}

extern "C" void kernel_launch(void* const* d_in, const int* in_sizes, int n_in,
                              void* d_out, int out_size, void* d_ws, size_t ws_size,
                              hipStream_t stream) {
  (void)n_in; (void)out_size; (void)d_ws; (void)ws_size;
  const int* idx        = (const int*)d_in[0];    // [B,H,W,K] integer
  const float* dists2   = (const float*)d_in[1];  // [B,H,W,K] f32
  const float* features = (const float*)d_in[2];  // [P,C]     f32
  const float* radius   = (const float*)d_in[3];  // [1]       f32
  float* out            = (float*)d_out;          // [B,H,W,C] f32

  const int npix = in_sizes[1] / KFRAG;           // B*H*W
  const int npts = in_sizes[2] / CFEAT;           // P

  const int ntiles = npix / TILE;
  int blocks = 2048;                               // persistent-ish, 4 tiles each
  if (ntiles > 0 && ntiles < blocks) blocks = ntiles;
  if (blocks < 1) blocks = 1;

  points_alpha_composite<<<dim3(blocks), dim3(TILE), 0, stream>>>(
      idx, dists2, features, radius, out, npix, npts);
}